// Warping_85529978732898
// MI455X (gfx1250) — compile-verified
//
#include <hip/hip_runtime.h>

// Warp/trilinear-resample kernel for MI455X (gfx1250, wave32).
//
// Layout (C-order, matches JAX):
//   ddf  : [B=4][160][160][160][3] f32
//   image: [B=4][160][160][160]    f32
//   out  : [B=4][160][160][160]    f32
//
// Roofline: ~330 MB HBM traffic @ 23.3 TB/s ~ 14 us; ~0.4 GFLOP total.
// Pure data-movement kernel -> no WMMA applicable (no contraction).
//
// Strategy:
//  - image (65.5 MB) kept L2-resident (192 MB L2): gathers use default RT hint.
//  - ddf / out streamed with non-temporal hints (TH_LOAD_NT / TH_STORE_NT).
//  - z-neighbors contiguous -> b64 pair loads: 4 gathers instead of 8.
//  - 3D grid (y, x, b) + block = 160 threads (5 wave32) along z: no integer
//    division in the decode, ddf/out accesses fully contiguous per wave.

#define DD1 160
#define DD2 160
#define DD3 160
#define NBATCH 4

typedef float f2 __attribute__((ext_vector_type(2)));
typedef f2 __attribute__((aligned(4))) f2_a4;   // allow dword-aligned b64 load

__global__ __launch_bounds__(DD3) void warp_trilinear_kernel(
    const float* __restrict__ ddf,
    const float* __restrict__ image,
    float* __restrict__ out)
{
    const int z = threadIdx.x;            // 0..159, stride-1 dim
    const int y = blockIdx.x;             // 0..159
    const int x = blockIdx.y;             // 0..159
    const int b = blockIdx.z;             // 0..3

    // Flat voxel index: ((b*160 + x)*160 + y)*160 + z  -- two scalar MADs.
    const int    row = (b * DD1 + x) * DD2 + y;
    const size_t vox = (size_t)row * DD3 + (size_t)z;

    // ---- streaming read of displacement (non-temporal: read-once) ----
    const float* dp = ddf + vox * 3u;
    const float dx = __builtin_nontemporal_load(dp + 0);
    const float dy = __builtin_nontemporal_load(dp + 1);
    const float dz = __builtin_nontemporal_load(dp + 2);

    // ---- warped coordinate, clamped to volume bounds (matches jnp.clip) ----
    float fx = fminf(fmaxf((float)x + dx, 0.0f), (float)(DD1 - 1));
    float fy = fminf(fmaxf((float)y + dy, 0.0f), (float)(DD2 - 1));
    float fz = fminf(fmaxf((float)z + dz, 0.0f), (float)(DD3 - 1));

    const float flx = floorf(fx), fly = floorf(fy), flz = floorf(fz);
    const float wx = fx - flx,  wy = fy - fly,  wz = fz - flz;
    const float ux = 1.0f - wx, uy = 1.0f - wy, uz = 1.0f - wz;

    const int x0 = (int)flx, y0 = (int)fly, z0 = (int)flz;
    const int x1 = min(x0 + 1, DD1 - 1);
    const int y1 = min(y0 + 1, DD2 - 1);

    // z-pair trick: load [zb, zb+1] in one b64; zb = min(z0, D-2).
    // If z0 == 159 (only when fz clamps to exactly 159.0, so wz == 0),
    // both v(z0) and v(z1) come from the .y element -> still exact.
    const int  zb = min(z0, DD3 - 2);
    const bool hi = (z0 != zb);

    const float* imgb = image + (size_t)b * (size_t)(DD1 * DD2 * DD3);
    const float* r00 = imgb + (size_t)((x0 * DD2 + y0) * DD3 + zb);
    const float* r01 = imgb + (size_t)((x0 * DD2 + y1) * DD3 + zb);
    const float* r10 = imgb + (size_t)((x1 * DD2 + y0) * DD3 + zb);
    const float* r11 = imgb + (size_t)((x1 * DD2 + y1) * DD3 + zb);

    // ---- 4 x global_load_b64 gathers (L2-resident working set) ----
    const f2 p00 = *(const f2_a4*)r00;
    const f2 p01 = *(const f2_a4*)r01;
    const f2 p10 = *(const f2_a4*)r10;
    const f2 p11 = *(const f2_a4*)r11;

    const float v000 = hi ? p00.y : p00.x, v001 = p00.y;
    const float v010 = hi ? p01.y : p01.x, v011 = p01.y;
    const float v100 = hi ? p10.y : p10.x, v101 = p10.y;
    const float v110 = hi ? p11.y : p11.x, v111 = p11.y;

    // ---- trilinear blend (same weight association as reference) ----
    const float c00 = v000 * uz + v001 * wz;
    const float c01 = v010 * uz + v011 * wz;
    const float c10 = v100 * uz + v101 * wz;
    const float c11 = v110 * uz + v111 * wz;
    const float c0  = c00 * uy + c01 * wy;
    const float c1  = c10 * uy + c11 * wy;
    const float res = c0 * ux + c1 * wx;

    // ---- streaming write (non-temporal: write-once) ----
    __builtin_nontemporal_store(res, out + vox);
}

extern "C" void kernel_launch(void* const* d_in, const int* in_sizes, int n_in,
                              void* d_out, int out_size, void* d_ws, size_t ws_size,
                              hipStream_t stream) {
    (void)in_sizes; (void)n_in; (void)out_size; (void)d_ws; (void)ws_size;
    const float* ddf   = (const float*)d_in[0];   // [4,160,160,160,3] f32
    const float* image = (const float*)d_in[1];   // [4,160,160,160]   f32
    float*       out   = (float*)d_out;           // [4,160,160,160]   f32

    // grid: x = y-rows, y = x-planes, z = batch; block = z-dim (5 wave32s)
    warp_trilinear_kernel<<<dim3(DD2, DD1, NBATCH), dim3(DD3), 0, stream>>>(ddf, image, out);
}